// GCN_res_17772574671069
// MI455X (gfx1250) — compile-verified
//
#include <hip/hip_runtime.h>
#include <hip/hip_bf16.h>

// ---------------------------------------------------------------------------
// GCN_res forward for MI455X (gfx1250, wave32).
// Dense GEMMs use V_WMMA_F32_16X16X4_F32 (full fp32 precision, matches the
// fp32 reference). Ncols is a template constant so every B load folds into a
// global_load_b32 immediate offset; each wave runs two independent WMMA
// accumulation chains (2 M-tiles) off a single shared B fragment.
// SpMM is an L2-resident float4 atomic scatter (h/t are 51 MB each -> both
// live in the 192 MB L2).
// ---------------------------------------------------------------------------

typedef __attribute__((ext_vector_type(2))) float v2f;
typedef __attribute__((ext_vector_type(8))) float v8f;

#define NODES 50000
#define HDIM  256
#define FDIM  128
#define CDIM  112
#define NLAYERS 8
#define BN_EPS 1e-5f

// ---------------------------- small helpers --------------------------------

__global__ void fill_f32(float* __restrict__ p, float v, long long n) {
    long long i = (long long)blockIdx.x * blockDim.x + threadIdx.x;
    if (i < n) p[i] = v;
}

// deg[] pre-filled with 1.0 (self loop); add 1 per incoming edge (col).
__global__ void deg_count(const int* __restrict__ ecol, float* __restrict__ deg, int E) {
    int e = blockIdx.x * blockDim.x + threadIdx.x;
    if (e < E) atomicAdd(&deg[ecol[e]], 1.0f);
}

// in-place deg -> rsqrt(deg)   (deg >= 1 always due to self loops)
__global__ void make_dinv(float* __restrict__ deg, int n) {
    int i = blockIdx.x * blockDim.x + threadIdx.x;
    if (i < n) deg[i] = rsqrtf(deg[i]);
}

// norm[e] = dinv[row]*dinv[col] for edges; dinv[n]^2 for self loops.
__global__ void compute_norm(const int* __restrict__ erow, const int* __restrict__ ecol,
                             const float* __restrict__ dinv, float* __restrict__ nrm,
                             int E, int n) {
    int e = blockIdx.x * blockDim.x + threadIdx.x;
    if (e < E) {
        nrm[e] = dinv[erow[e]] * dinv[ecol[e]];
    } else if (e < E + n) {
        float d = dinv[e - E];
        nrm[e] = d * d;
    }
}

__global__ void layerw_softmax(const float* __restrict__ lw, float* __restrict__ w) {
    if (threadIdx.x == 0) {
        float mx = lw[0];
        for (int i = 1; i < NLAYERS; ++i) mx = fmaxf(mx, lw[i]);
        float s = 0.f;
        float e[NLAYERS];
        for (int i = 0; i < NLAYERS; ++i) { e[i] = expf(lw[i] - mx); s += e[i]; }
        for (int i = 0; i < NLAYERS; ++i) w[i] = e[i] / s;
    }
}

// ------------------------- WMMA fp32 GEMM ----------------------------------
// C[M x NC] = A[M x K] @ W[K x NC] + bias.
// Block = 128 threads (4 waves). blockIdx.x -> 32-row A tile (staged in LDS,
// shared by all 4 waves). blockIdx.y*4 + wave -> 16-col tile. Each wave keeps
// TWO 16x16 fp32 accumulators (rows m0..m0+15 and m0+16..m0+31) fed by one
// shared B fragment per k-step: half the global loads, two independent
// WMMA chains to hide the XDL RAW latency.
// Fragment layouts per CDNA5 ISA 7.12.2 (32-bit A 16x4 / B 4x16 / C 16x16).
template <int K, int NC>
__global__ __launch_bounds__(128)
void wmma_gemm_bias(const float* __restrict__ A, const float* __restrict__ W,
                    const float* __restrict__ bias, float* __restrict__ Cout,
                    int M) {
    __shared__ float As[32 * K];
    const int tid  = threadIdx.x;
    const int lane = tid & 31;
    const int wave = tid >> 5;
    const int m0   = blockIdx.x << 5;

    // stage the 32 x K A-tile (coalesced; shared by the 4 waves).
    // Row index clamped so the final (partial) block stays in bounds.
    for (int idx = tid; idx < 32 * K; idx += 128) {
        int r = idx / K, c = idx % K;               // K is a power of two
        int row = m0 + r;
        if (row >= M) row = M - 1;
        As[idx] = A[(long long)row * K + c];
    }
    __syncthreads();

    const int n0 = ((blockIdx.y << 2) + wave) << 4;
    if (n0 >= NC) return;                           // wave-uniform exit

    const int half = lane >> 4;                     // 0: lanes 0-15, 1: 16-31
    const int l16  = lane & 15;
    const int bcol = n0 + l16;

    // per-lane bases: all k-loop offsets below are compile-time constants,
    // folding straight into global_load/ds_load immediate offsets.
    const float* bptr = W + (size_t)(2 * half) * NC + bcol;  // B rows 2*half, +1
    const float* a0 = &As[l16 * K];                          // tile 0 row
    const float* a1 = &As[(16 + l16) * K];                   // tile 1 row

    v8f c0 = {}, c1 = {};
    #pragma unroll
    for (int k = 0; k < K; k += 4) {
        const int ka = k + 2 * half;                // half 0 -> K0,K1; half 1 -> K2,K3
        v2f b, x0, x1;
        b.x  = bptr[(size_t)k * NC];                // W[(k+2*half)  *NC + bcol]
        b.y  = bptr[(size_t)k * NC + NC];           // W[(k+2*half+1)*NC + bcol]
        x0.x = a0[ka];  x0.y = a0[ka + 1];
        x1.x = a1[ka];  x1.y = a1[ka + 1];
        c0 = __builtin_amdgcn_wmma_f32_16x16x4_f32(false, x0, false, b,
                                                   (short)0, c0, false, false);
        c1 = __builtin_amdgcn_wmma_f32_16x16x4_f32(false, x1, false, b,
                                                   (short)0, c1, false, false);
    }

    const float bv = bias[bcol];
    const int rbase = m0 + 8 * half;                // VGPR r: lanes0-15 M=r, lanes16-31 M=r+8
    #pragma unroll
    for (int r = 0; r < 8; ++r) {
        const int row0 = rbase + r;
        const int row1 = row0 + 16;
        if (row0 < M) Cout[(long long)row0 * NC + bcol] = c0[r] + bv;
        if (row1 < M) Cout[(long long)row1 * NC + bcol] = c1[r] + bv;
    }
}

// --------------------------- SpMM scatter ----------------------------------
// 64 consecutive threads cover one 256-float row (float4 each): coalesced
// 1 KB reads of t[row], coalesced atomic adds into hout[col]. Self loops are
// the tail range e in [E, E+N).
__global__ void spmm_scatter(const float* __restrict__ t, const int* __restrict__ erow,
                             const int* __restrict__ ecol, const float* __restrict__ nrm,
                             float* __restrict__ hout, int E, int n) {
    long long tid = (long long)blockIdx.x * blockDim.x + threadIdx.x;
    long long total = (long long)(E + n) * (HDIM / 4);
    if (tid >= total) return;
    int e  = (int)(tid >> 6);
    int f4 = ((int)tid & 63) << 2;
    int row, col;
    if (e < E) { row = erow[e]; col = ecol[e]; }
    else       { row = e - E;  col = row;     }
    const float s = nrm[e];
    const float4 v = *(const float4*)(t + (long long)row * HDIM + f4);
    float* dst = hout + (long long)col * HDIM + f4;
    atomicAdd(dst + 0, v.x * s);
    atomicAdd(dst + 1, v.y * s);
    atomicAdd(dst + 2, v.z * s);
    atomicAdd(dst + 3, v.w * s);
}

// --------------------------- BatchNorm -------------------------------------
// sums[0..H) = sum, sums[H..2H) = sum of squares (column-wise over nodes).
__global__ void bn_stats(const float* __restrict__ h, float* __restrict__ sums,
                         int n, int rowsPerBlock) {
    const int f  = threadIdx.x;                 // blockDim.x == HDIM
    const int r0 = blockIdx.x * rowsPerBlock;
    const int r1 = min(r0 + rowsPerBlock, n);
    float s = 0.f, s2 = 0.f;
    for (int r = r0; r < r1; ++r) {
        float v = h[(long long)r * HDIM + f];
        s += v; s2 += v * v;
    }
    atomicAdd(&sums[f], s);
    atomicAdd(&sums[HDIM + f], s2);
}

// h = relu((h-mu)*rsqrt(var+eps)*gamma+beta);  acc += w[layer]*h
__global__ void bn_apply_relu_res(float* __restrict__ h, float* __restrict__ acc,
                                  const float* __restrict__ sums,
                                  const float* __restrict__ gamma,
                                  const float* __restrict__ beta,
                                  const float* __restrict__ wsm, int layer, int n) {
    long long i = (long long)blockIdx.x * blockDim.x + threadIdx.x;
    if (i >= (long long)n * HDIM) return;
    int f = (int)(i & (HDIM - 1));
    float inv_n = 1.0f / (float)n;
    float mu  = sums[f] * inv_n;
    float var = sums[HDIM + f] * inv_n - mu * mu;
    float v = (h[i] - mu) * rsqrtf(var + BN_EPS) * gamma[f] + beta[f];
    v = fmaxf(v, 0.f);
    h[i] = v;
    acc[i] += wsm[layer] * v;
}

// --------------------------- log-softmax -----------------------------------
// one wave32 per row of C=112 values, in place on d_out.
__global__ void row_log_softmax(float* __restrict__ out, int n) {
    int row  = blockIdx.x * (blockDim.x >> 5) + (threadIdx.x >> 5);
    int lane = threadIdx.x & 31;
    if (row >= n) return;
    float* p = out + (long long)row * CDIM;
    float m = -3.4e38f;
    for (int f = lane; f < CDIM; f += 32) m = fmaxf(m, p[f]);
    for (int o = 16; o > 0; o >>= 1) m = fmaxf(m, __shfl_xor(m, o));
    float s = 0.f;
    for (int f = lane; f < CDIM; f += 32) s += expf(p[f] - m);
    for (int o = 16; o > 0; o >>= 1) s += __shfl_xor(s, o);
    float ls = logf(s) + m;
    for (int f = lane; f < CDIM; f += 32) p[f] -= ls;
}

// ---------------------------------------------------------------------------

extern "C" void kernel_launch(void* const* d_in, const int* in_sizes, int n_in,
                              void* d_out, int out_size, void* d_ws, size_t ws_size,
                              hipStream_t stream) {
    const int N = NODES, H = HDIM, E = in_sizes[1] / 2;

    const float* x      = (const float*)d_in[0];
    const int*   edge1  = (const int*)  d_in[1];   // [2,E] flat: rows, then cols
    const int*   edge2  = (const int*)  d_in[2];
    const float* in_W   = (const float*)d_in[3];
    const float* in_b   = (const float*)d_in[4];
    const float* convW  = (const float*)d_in[5];   // [L,H,H]
    const float* convB  = (const float*)d_in[6];   // [L,H]
    const float* bn_g   = (const float*)d_in[7];
    const float* bn_b   = (const float*)d_in[8];
    const float* out_W  = (const float*)d_in[9];
    const float* out_b  = (const float*)d_in[10];
    const float* lw     = (const float*)d_in[11];
    float* out = (float*)d_out;

    // ---- workspace layout (floats); total ~161 MB ----
    float* ws    = (float*)d_ws;
    float* dinv1 = ws;  ws += N;
    float* dinv2 = ws;  ws += N;
    float* wsm   = ws;  ws += 16;                  // softmax(layer_w), padded
    float* sums  = ws;  ws += 2 * H;
    float* norm1 = ws;  ws += (size_t)(E + N);
    float* norm2 = ws;  ws += (size_t)(E + N);
    float* h     = ws;  ws += (size_t)N * H;
    float* t     = ws;  ws += (size_t)N * H;
    float* acc   = ws;

    const long long NH = (long long)N * H;
    const unsigned mblocks = (unsigned)((N + 31) / 32);

    // ---- residual weights + GCN normalization ----
    layerw_softmax<<<1, 32, 0, stream>>>(lw, wsm);
    fill_f32<<<(N + 255) / 256, 256, 0, stream>>>(dinv1, 1.0f, N);
    fill_f32<<<(N + 255) / 256, 256, 0, stream>>>(dinv2, 1.0f, N);
    deg_count<<<(E + 255) / 256, 256, 0, stream>>>(edge1 + E, dinv1, E);
    deg_count<<<(E + 255) / 256, 256, 0, stream>>>(edge2 + E, dinv2, E);
    make_dinv<<<(N + 255) / 256, 256, 0, stream>>>(dinv1, N);
    make_dinv<<<(N + 255) / 256, 256, 0, stream>>>(dinv2, N);
    compute_norm<<<(E + N + 255) / 256, 256, 0, stream>>>(edge1, edge1 + E, dinv1, norm1, E, N);
    compute_norm<<<(E + N + 255) / 256, 256, 0, stream>>>(edge2, edge2 + E, dinv2, norm2, E, N);

    // ---- acc = 0 ; h = x @ in_W + in_b ----
    fill_f32<<<(unsigned)((NH + 255) / 256), 256, 0, stream>>>(acc, 0.f, NH);
    wmma_gemm_bias<FDIM, HDIM><<<dim3(mblocks, (H + 63) / 64), 128, 0, stream>>>(
        x, in_W, in_b, h, N);

    // ---- 8 GCN layers ----
    const long long scat_total = (long long)(E + N) * (H / 4);
    for (int i = 0; i < NLAYERS; ++i) {
        const int*   erow = (i < NLAYERS / 2) ? edge1 : edge2;
        const float* nrm  = (i < NLAYERS / 2) ? norm1 : norm2;

        // t = h @ convW[i] + convB[i]
        wmma_gemm_bias<HDIM, HDIM><<<dim3(mblocks, (H + 63) / 64), 128, 0, stream>>>(
            h, convW + (size_t)i * H * H, convB + (size_t)i * H, t, N);

        // h = scatter-add( t[row] * norm  ->  col )
        fill_f32<<<(unsigned)((NH + 255) / 256), 256, 0, stream>>>(h, 0.f, NH);
        spmm_scatter<<<(unsigned)((scat_total + 255) / 256), 256, 0, stream>>>(
            t, erow, erow + E, nrm, h, E, N);

        // BatchNorm (batch stats) + ReLU + weighted residual accumulate
        fill_f32<<<(2 * H + 255) / 256, 256, 0, stream>>>(sums, 0.f, 2 * H);
        bn_stats<<<(N + 63) / 64, H, 0, stream>>>(h, sums, N, 64);
        bn_apply_relu_res<<<(unsigned)((NH + 255) / 256), 256, 0, stream>>>(
            h, acc, sums, bn_g + (size_t)i * H, bn_b + (size_t)i * H, wsm, i, N);
    }

    // ---- out = acc @ out_W + out_b ; log_softmax rows ----
    wmma_gemm_bias<HDIM, CDIM><<<dim3(mblocks, (CDIM + 63) / 64), 128, 0, stream>>>(
        acc, out_W, out_b, out, N);
    row_log_softmax<<<(N + 7) / 8, 256, 0, stream>>>(out, N);
}